// spatialGCN_33346126086836
// MI455X (gfx1250) — compile-verified
//
#include <hip/hip_runtime.h>
#include <hip/hip_bf16.h>
#include <math.h>

// ---------------------------------------------------------------------------
// spatialGCN for MI455X (gfx1250, wave32).
// Strategy: pad all GEMM operands to 16-multiples once (zero padding on the
// adjacency/weight side makes padded lanes contribute exactly 0), so every
// WMMA K-loop is branch-free with aligned b64 fragment loads. All matmul
// FLOPs go through V_WMMA_F32_16X16X4_F32 (full f32 precision).
// ---------------------------------------------------------------------------

typedef float v2f __attribute__((ext_vector_type(2)));
typedef float v8f __attribute__((ext_vector_type(8)));

static constexpr int Bn   = 512;  // batch
static constexpr int Nn   = 207;  // nodes
static constexpr int Cc   = 207;  // channels
static constexpr int NP   = 208;  // padded nodes/channels (13*16)
static constexpr int KP3  = 624;  // padded 3C (3*208)
static constexpr int COUT = 512;
static constexpr float EPS = 1e-5f;

__device__ __forceinline__ v8f wmma4(v2f a, v2f b, v8f c) {
  // (neg_a, A, neg_b, B, c_mod, C, reuse_a, reuse_b)
  return __builtin_amdgcn_wmma_f32_16x16x4_f32(false, a, false, b, (short)0, c,
                                               false, false);
}

// --------------------------- adjacency (padded) ----------------------------
// adp_pad[w,v] = softmax(relu(E1@E2), axis=1), 208x208, pad row/col zeroed.
__global__ void adj_softmax_kernel(const float* __restrict__ E1,
                                   const float* __restrict__ E2,
                                   float* __restrict__ adp) {
  __shared__ float sdata[256];
  const int w = blockIdx.x;  // 0..207
  const int v = threadIdx.x;
  if (w == Nn) {  // padding row: all zeros (block-uniform branch)
    if (v < NP) adp[w * NP + v] = 0.f;
    return;
  }
  float val = -1e30f;
  if (v < Nn) {
    float dot = 0.f;
#pragma unroll
    for (int k = 0; k < 10; ++k) dot += E1[w * 10 + k] * E2[k * Nn + v];
    val = dot > 0.f ? dot : 0.f;  // relu
  }
  sdata[threadIdx.x] = val;
  __syncthreads();
  for (int s = 128; s > 0; s >>= 1) {
    if (threadIdx.x < s)
      sdata[threadIdx.x] = fmaxf(sdata[threadIdx.x], sdata[threadIdx.x + s]);
    __syncthreads();
  }
  const float m = sdata[0];
  __syncthreads();
  const float e = (v < Nn) ? expf(val - m) : 0.f;
  sdata[threadIdx.x] = e;
  __syncthreads();
  for (int s = 128; s > 0; s >>= 1) {
    if (threadIdx.x < s) sdata[threadIdx.x] += sdata[threadIdx.x + s];
    __syncthreads();
  }
  if (v < NP) adp[w * NP + v] = (v < Nn) ? e / sdata[0] : 0.f;
}

// --------------------------- packing kernels -------------------------------
__global__ void pack_x_kernel(const float* __restrict__ x, float* __restrict__ xp) {
  const int idx = blockIdx.x * 256 + threadIdx.x;  // covers 512*208*208
  const int b = idx / (NP * NP);
  const int rem = idx - b * NP * NP;
  const int n = rem / NP, c = rem - n * NP;
  xp[idx] = (n < Nn && c < Cc) ? x[((size_t)b * Nn + n) * Cc + c] : 0.f;
}

__global__ void pack_wg_kernel(const float* __restrict__ Wg, float* __restrict__ Wgp) {
  const int idx = blockIdx.x * 256 + threadIdx.x;  // covers 2*208*624
  const int l = idx / (NP * KP3);
  const int rem = idx - l * NP * KP3;
  const int o = rem / KP3, k = rem - o * KP3;
  const int s = k / NP, c = k - s * NP;
  Wgp[idx] = (o < Cc && c < Cc)
                 ? Wg[((size_t)l * Cc + o) * (3 * Cc) + s * Cc + c]
                 : 0.f;
}

__global__ void pack_wm_kernel(const float* __restrict__ Wm, float* __restrict__ Wmp) {
  const int idx = blockIdx.x * 256 + threadIdx.x;  // covers 512*208
  const int o = idx / NP, c = idx - o * NP;
  Wmp[idx] = (c < Cc) ? Wm[o * Cc + c] : 0.f;
}

// --------------------------- diffusion (padded) ----------------------------
// D[b] = adp(208x208) @ H[b](208x208); one wave per 16x16 tile, branch-free.
__global__ void diffusion_kernel(const float* __restrict__ A,
                                 const float* __restrict__ H,
                                 float* __restrict__ D, int totalWaves) {
  const int wid = blockIdx.x * (blockDim.x >> 5) + (threadIdx.x >> 5);
  if (wid >= totalWaves) return;
  const int b  = wid / 169;
  const int t  = wid - b * 169;
  const int mt = t / 13;
  const int nt = t - mt * 13;

  const float* __restrict__ Hb = H + (size_t)b * NP * NP;
  float* __restrict__ Db       = D + (size_t)b * NP * NP;

  const int lane = threadIdx.x & 31;
  const int half = lane >> 4;  // which K pair this lane holds
  const int r    = lane & 15;

  const float* Ap = A + (mt * 16 + r) * NP + half * 2;   // A row, b64 chunks
  const float* Bp = Hb + half * 2 * NP + nt * 16 + r;    // B column

  v8f acc = {};
#pragma unroll 4
  for (int k0 = 0; k0 < NP; k0 += 4) {
    v2f a = *(const v2f*)Ap;
    v2f bf;
    bf.x = Bp[0];
    bf.y = Bp[NP];
    acc  = wmma4(a, bf, acc);
    Ap += 4;
    Bp += 4 * NP;
  }
#pragma unroll
  for (int j = 0; j < 8; ++j)
    Db[(mt * 16 + j + half * 8) * NP + nt * 16 + r] = acc[j];
}

// --------------------------- GCN 1x1 conv + BN -----------------------------
// Hout[t,o] = BN( sum_k [H|X1|X2][t,k] * Wgp[o,k] + bg[o] ).
// 4 M-tiles per wave (64 rows), K split into 3 uniform 208-wide slices.
__global__ void gcn_kernel(const float* __restrict__ H,
                           const float* __restrict__ X1,
                           const float* __restrict__ X2,
                           const float* __restrict__ Wgp,  // (208,624) padded
                           const float* __restrict__ bg,   // (207)
                           const float* __restrict__ gamma,
                           const float* __restrict__ beta,
                           const float* __restrict__ rmean,
                           const float* __restrict__ rvar,
                           float* __restrict__ Hout, int totalWaves) {
  const int wid = blockIdx.x * (blockDim.x >> 5) + (threadIdx.x >> 5);
  if (wid >= totalWaves) return;
  const int mb = wid / 13;       // M block of 4 tiles (1664 blocks, exact)
  const int nt = wid - mb * 13;  // channel tile

  const int lane = threadIdx.x & 31;
  const int half = lane >> 4;
  const int r    = lane & 15;
  const int rowb = mb * 64 + r;       // lane's row in tile 0
  const int col  = nt * 16 + r;       // 0..207, in padded range

  v8f acc0 = {}, acc1 = {}, acc2 = {}, acc3 = {};
  const float* Bp = Wgp + (size_t)col * KP3 + half * 2;  // contiguous in k
  const float* srcs[3] = {H, X1, X2};
#pragma unroll
  for (int s = 0; s < 3; ++s) {
    const float* A0 = srcs[s] + (size_t)rowb * NP + half * 2;
#pragma unroll 2
    for (int k0 = 0; k0 < NP; k0 += 4) {
      v2f bf = *(const v2f*)Bp;
      v2f a0 = *(const v2f*)A0;
      v2f a1 = *(const v2f*)(A0 + 16 * NP);
      v2f a2 = *(const v2f*)(A0 + 32 * NP);
      v2f a3 = *(const v2f*)(A0 + 48 * NP);
      acc0 = wmma4(a0, bf, acc0);
      acc1 = wmma4(a1, bf, acc1);
      acc2 = wmma4(a2, bf, acc2);
      acc3 = wmma4(a3, bf, acc3);
      A0 += 4;
      Bp += 4;
    }
  }
  // BN epilogue (clamp col for the one padded channel; its output is unused)
  const int cc    = (col < Cc) ? col : 0;
  const float inv   = gamma[cc] / sqrtf(rvar[cc] + EPS);
  const float shift = beta[cc] - rmean[cc] * inv;
  const float bias  = bg[cc];
  v8f accs[4] = {acc0, acc1, acc2, acc3};
#pragma unroll
  for (int i = 0; i < 4; ++i)
#pragma unroll
    for (int j = 0; j < 8; ++j) {
      const int m = mb * 64 + i * 16 + j + half * 8;
      Hout[(size_t)m * NP + col] = (accs[i][j] + bias) * inv + shift;
    }
}

// --------------------------- final 1x1 conv (C -> 512) ---------------------
__global__ void final_kernel(const float* __restrict__ H,
                             const float* __restrict__ Wmp,  // (512,208) padded
                             const float* __restrict__ bm,   // (512)
                             float* __restrict__ Out, int totalWaves) {
  const int wid = blockIdx.x * (blockDim.x >> 5) + (threadIdx.x >> 5);
  if (wid >= totalWaves) return;
  const int mb = wid / 32;       // M block of 4 tiles
  const int nt = wid - mb * 32;  // out-channel tile (512 = 32*16 exact)

  const int lane = threadIdx.x & 31;
  const int half = lane >> 4;
  const int r    = lane & 15;
  const int rowb = mb * 64 + r;
  const int col  = nt * 16 + r;

  v8f acc0 = {}, acc1 = {}, acc2 = {}, acc3 = {};
  const float* Bp = Wmp + (size_t)col * NP + half * 2;
  const float* A0 = H + (size_t)rowb * NP + half * 2;
#pragma unroll 2
  for (int k0 = 0; k0 < NP; k0 += 4) {
    v2f bf = *(const v2f*)Bp;
    v2f a0 = *(const v2f*)A0;
    v2f a1 = *(const v2f*)(A0 + 16 * NP);
    v2f a2 = *(const v2f*)(A0 + 32 * NP);
    v2f a3 = *(const v2f*)(A0 + 48 * NP);
    acc0 = wmma4(a0, bf, acc0);
    acc1 = wmma4(a1, bf, acc1);
    acc2 = wmma4(a2, bf, acc2);
    acc3 = wmma4(a3, bf, acc3);
    A0 += 4;
    Bp += 4;
  }
  const float bias = bm[col];
  v8f accs[4] = {acc0, acc1, acc2, acc3};
#pragma unroll
  for (int i = 0; i < 4; ++i)
#pragma unroll
    for (int j = 0; j < 8; ++j) {
      const int m = mb * 64 + i * 16 + j + half * 8;  // padded token id
      const int b = m / NP;
      const int n = m - b * NP;
      if (n < Nn)  // drop the padded token row (store-side only)
        Out[((size_t)b * Nn + n) * COUT + col] = accs[i][j] + bias;
    }
}

// ---------------------------------------------------------------------------
extern "C" void kernel_launch(void* const* d_in, const int* in_sizes, int n_in,
                              void* d_out, int out_size, void* d_ws,
                              size_t ws_size, hipStream_t stream) {
  const float* x     = (const float*)d_in[0];   // (B,N,C) token-major
  const float* e1    = (const float*)d_in[1];   // (N,10)
  const float* e2    = (const float*)d_in[2];   // (10,N)
  const float* Wg    = (const float*)d_in[3];   // (L,C,3C)
  const float* bg    = (const float*)d_in[4];   // (L,C)
  const float* gamma = (const float*)d_in[5];
  const float* beta  = (const float*)d_in[6];
  const float* rmean = (const float*)d_in[7];
  const float* rvar  = (const float*)d_in[8];
  const float* Wm    = (const float*)d_in[9];   // (512,C)
  const float* bm    = (const float*)d_in[10];  // (512)
  float* out = (float*)d_out;

  const size_t featP = (size_t)Bn * NP * NP;  // 22,151,168 floats per buffer

  // Workspace (floats): adp | Wgp | Wmp | x0 | x1 | x2 | h1  (~356 MB)
  // h2 reuses the x0 buffer (x0 is dead after the layer-0 GCN).
  float* ws   = (float*)d_ws;
  float* adp  = ws;                       //  43,264
  float* Wgp  = adp + 43264;              // 259,584
  float* Wmp  = Wgp + 259584;             // 106,496
  float* x0   = Wmp + 106496 + 192;       // align to 64 floats
  float* x1   = x0 + featP;
  float* x2   = x1 + featP;
  float* h1   = x2 + featP;
  float* h2   = x0;  // alias (safe: layer-1 GCN reads h1/x1/x2 only)

  const dim3 blk(256);  // 8 waves per block
  const int WPB = 8;

  // ---- packing / adjacency ----
  adj_softmax_kernel<<<NP, blk, 0, stream>>>(e1, e2, adp);
  pack_x_kernel<<<(int)(featP / 256), blk, 0, stream>>>(x, x0);         // 86528
  pack_wg_kernel<<<(2 * NP * KP3) / 256, blk, 0, stream>>>(Wg, Wgp);    // 1014
  pack_wm_kernel<<<(COUT * NP) / 256, blk, 0, stream>>>(Wm, Wmp);       // 416

  const int diffWaves = Bn * 169;          // 86528  (/8 exact)
  const int gcnWaves  = 1664 * 13;         // 21632  (/8 exact)
  const int finWaves  = 1664 * 32;         // 53248  (/8 exact)

  // ---- layer 0 ----
  diffusion_kernel<<<diffWaves / WPB, blk, 0, stream>>>(adp, x0, x1, diffWaves);
  diffusion_kernel<<<diffWaves / WPB, blk, 0, stream>>>(adp, x1, x2, diffWaves);
  gcn_kernel<<<gcnWaves / WPB, blk, 0, stream>>>(
      x0, x1, x2, Wgp, bg, gamma, beta, rmean, rvar, h1, gcnWaves);

  // ---- layer 1 ----
  diffusion_kernel<<<diffWaves / WPB, blk, 0, stream>>>(adp, h1, x1, diffWaves);
  diffusion_kernel<<<diffWaves / WPB, blk, 0, stream>>>(adp, x1, x2, diffWaves);
  gcn_kernel<<<gcnWaves / WPB, blk, 0, stream>>>(
      h1, x1, x2, Wgp + (size_t)NP * KP3, bg + Cc, gamma + Cc, beta + Cc,
      rmean + Cc, rvar + Cc, h2, gcnWaves);

  // ---- final 1x1 conv ----
  final_kernel<<<finWaves / WPB, blk, 0, stream>>>(h2, Wmp, bm, out, finWaves);
}